// SGAHead_54391465836643
// MI455X (gfx1250) — compile-verified
//
#include <hip/hip_runtime.h>

typedef __attribute__((ext_vector_type(2))) float v2f;
typedef __attribute__((ext_vector_type(8))) float v8f;

#define WMMA_F32(A, Bm, Cm) \
  __builtin_amdgcn_wmma_f32_16x16x4_f32(false, (A), false, (Bm), (short)0, (Cm), false, false)

static constexpr int Bc = 8, Cc = 256, Hc = 128, Wc = 128;
static constexpr int HWc = Hc * Wc;                       // 16384
static constexpr size_t OUT_N  = (size_t)Bc * Cc * Hc * Wc;   // 33,554,432
static constexpr size_t ATTH_N = (size_t)Bc * Wc * Hc * Hc;   // 16,777,216
static constexpr size_t ATTW_N = (size_t)Bc * Hc * Wc * Wc;   // 16,777,216

// ---------------------------------------------------------------------------
// CDNA5 async memory->LDS copies (ASYNCcnt path, no VGPR round trip).
// Flat shared pointers carry the LDS byte offset in addr[31:0].
// ---------------------------------------------------------------------------
__device__ __forceinline__ void async_ld_b32(const float* g, void* l) {
  unsigned lo = (unsigned)(size_t)l;
  unsigned long long ga = (unsigned long long)(size_t)g;
  asm volatile("global_load_async_to_lds_b32 %0, %1, off"
               :: "v"(lo), "v"(ga) : "memory");
}
__device__ __forceinline__ void async_ld_b128(const float* g, void* l) {
  unsigned lo = (unsigned)(size_t)l;
  unsigned long long ga = (unsigned long long)(size_t)g;
  asm volatile("global_load_async_to_lds_b128 %0, %1, off"
               :: "v"(lo), "v"(ga) : "memory");
}
__device__ __forceinline__ void wait_async0() {
  asm volatile("s_wait_asynccnt 0x0" ::: "memory");
}

// ---------------------------------------------------------------------------
// Kernel A: fused QKV projection.  Q = Wq*edge + bq, K = Wk*edge + bk,
// V = Wv*mask + bv.  GEMM: M=C=256, N=32 per block, K=C=256.
// Block = 256 threads (8 waves); wave wv owns M rows [wv*32, wv*32+32).
// ---------------------------------------------------------------------------
__global__ __launch_bounds__(256) void qkv_kernel(
    const float* __restrict__ edge, const float* __restrict__ maskf,
    const float* __restrict__ Wq, const float* __restrict__ bq,
    const float* __restrict__ Wk, const float* __restrict__ bk,
    const float* __restrict__ Wv, const float* __restrict__ bv,
    float* __restrict__ qo, float* __restrict__ ko, float* __restrict__ vo)
{
  __shared__ float xe[64][33];
  __shared__ float xm[64][33];

  const int t  = threadIdx.x;
  const int b  = blockIdx.y;
  const int n0 = blockIdx.x * 32;
  const int wv = t >> 5;
  const int tn = t & 15;
  const int hh = (t >> 4) & 1;     // half-wave (K sub-slot)
  const int mb = wv * 32;

  const v8f zero = {0.f, 0.f, 0.f, 0.f, 0.f, 0.f, 0.f, 0.f};
  v8f accQ[2][2], accK[2][2], accV[2][2];
#pragma unroll
  for (int i = 0; i < 2; i++)
#pragma unroll
    for (int j = 0; j < 2; j++) { accQ[i][j] = zero; accK[i][j] = zero; accV[i][j] = zero; }

  for (int kc = 0; kc < Cc; kc += 64) {
    __syncthreads();
#pragma unroll
    for (int i = 0; i < 8; i++) {
      int idx = i * 256 + t;
      int r = idx >> 5, cl = idx & 31;
      size_t g = (size_t)(b * Cc + kc + r) * HWc + n0 + cl;
      async_ld_b32(edge + g, &xe[r][cl]);
      async_ld_b32(maskf + g, &xm[r][cl]);
    }
    wait_async0();
    __syncthreads();

    for (int k = 0; k < 64; k += 4) {
      const int kg = kc + k + 2 * hh;
      v2f aq[2], ak[2], av[2], be[2], bm[2];
#pragma unroll
      for (int mt = 0; mt < 2; mt++) {
        int m = mb + mt * 16 + tn;
        aq[mt] = *(const v2f*)&Wq[m * Cc + kg];
        ak[mt] = *(const v2f*)&Wk[m * Cc + kg];
        av[mt] = *(const v2f*)&Wv[m * Cc + kg];
      }
#pragma unroll
      for (int nt = 0; nt < 2; nt++) {
        int n = nt * 16 + tn;
        be[nt].x = xe[k + 2 * hh][n];  be[nt].y = xe[k + 2 * hh + 1][n];
        bm[nt].x = xm[k + 2 * hh][n];  bm[nt].y = xm[k + 2 * hh + 1][n];
      }
#pragma unroll
      for (int mt = 0; mt < 2; mt++)
#pragma unroll
        for (int nt = 0; nt < 2; nt++) {
          accQ[mt][nt] = WMMA_F32(aq[mt], be[nt], accQ[mt][nt]);
          accK[mt][nt] = WMMA_F32(ak[mt], be[nt], accK[mt][nt]);
          accV[mt][nt] = WMMA_F32(av[mt], bm[nt], accV[mt][nt]);
        }
    }
  }

#pragma unroll
  for (int mt = 0; mt < 2; mt++)
#pragma unroll
    for (int nt = 0; nt < 2; nt++)
#pragma unroll
      for (int r = 0; r < 8; r++) {
        int row = mb + mt * 16 + r + 8 * hh;
        int col = n0 + nt * 16 + tn;
        size_t g = (size_t)(b * Cc + row) * HWc + col;
        qo[g] = accQ[mt][nt][r] + bq[row];
        ko[g] = accK[mt][nt][r] + bk[row];
        vo[g] = accV[mt][nt][r] + bv[row];
      }
}

// ---------------------------------------------------------------------------
// Kernel B: energies.  One block per (b, x) computes a 128x128 Gram matrix
// E[p][j] = sum_c q(c,p) * k(c,j) with element address
//   (b*C + c)*HW + x*sx + p*sp.
// E_H: x=w, (sx,sp)=(1,W).  E_W: x=h, (sx,sp)=(W,1).
// Raw energies land directly in the att output regions (softmaxed in place).
// ---------------------------------------------------------------------------
__global__ __launch_bounds__(256) void energy_kernel(
    const float* __restrict__ q, const float* __restrict__ kf,
    float* __restrict__ eout, int sx, int sp)
{
  __shared__ float qs[32][132];
  __shared__ float ks[32][132];

  const int t  = threadIdx.x;
  const int x  = blockIdx.x;
  const int b  = blockIdx.y;
  const int wv = t >> 5;
  const int tn = t & 15;
  const int hh = (t >> 4) & 1;
  const int wm = (wv >> 2) * 64;   // M offset: 0 / 64
  const int wn = (wv & 3) * 32;    // N offset: 0 / 32 / 64 / 96
  const size_t xoff = (size_t)x * sx;

  const v8f zero = {0.f, 0.f, 0.f, 0.f, 0.f, 0.f, 0.f, 0.f};
  v8f acc[4][2];
#pragma unroll
  for (int i = 0; i < 4; i++) { acc[i][0] = zero; acc[i][1] = zero; }

  for (int c0 = 0; c0 < Cc; c0 += 32) {
    __syncthreads();
#pragma unroll
    for (int i = 0; i < 16; i++) {
      int idx = i * 256 + t;
      int c = idx >> 7, p = idx & 127;
      size_t g = (size_t)(b * Cc + c0 + c) * HWc + xoff + (size_t)p * sp;
      async_ld_b32(q + g, &qs[c][p]);
      async_ld_b32(kf + g, &ks[c][p]);
    }
    wait_async0();
    __syncthreads();

    for (int k = 0; k < 32; k += 4) {
      v2f a[4], bb[2];
#pragma unroll
      for (int mt = 0; mt < 4; mt++) {
        int m = wm + mt * 16 + tn;
        a[mt].x = qs[k + 2 * hh][m];
        a[mt].y = qs[k + 2 * hh + 1][m];
      }
#pragma unroll
      for (int nt = 0; nt < 2; nt++) {
        int n = wn + nt * 16 + tn;
        bb[nt].x = ks[k + 2 * hh][n];
        bb[nt].y = ks[k + 2 * hh + 1][n];
      }
#pragma unroll
      for (int mt = 0; mt < 4; mt++)
#pragma unroll
        for (int nt = 0; nt < 2; nt++)
          acc[mt][nt] = WMMA_F32(a[mt], bb[nt], acc[mt][nt]);
    }
  }

  const size_t eb = ((size_t)b * 128 + x) * 16384;
#pragma unroll
  for (int mt = 0; mt < 4; mt++)
#pragma unroll
    for (int nt = 0; nt < 2; nt++)
#pragma unroll
      for (int r = 0; r < 8; r++) {
        int row = wm + mt * 16 + r + 8 * hh;
        int col = wn + nt * 16 + tn;
        eout[eb + (size_t)row * 128 + col] = acc[mt][nt][r];
      }
}

// ---------------------------------------------------------------------------
// Kernel C: joint softmax over the 256 concatenated logits per (b,h,w).
// One wave per position; float4 per lane covers each 128-float row.
// ---------------------------------------------------------------------------
__global__ __launch_bounds__(256) void softmax_kernel(
    float* __restrict__ attH, float* __restrict__ attW)
{
  const int t  = threadIdx.x;
  const int wv = t >> 5;
  const int l  = t & 31;
  const int p  = blockIdx.x * 8 + wv;     // b*H*W + h*W + w
  const int b  = p >> 14;
  const int h  = (p >> 7) & 127;
  const int w  = p & 127;

  float* rowH = attH + (((size_t)b * 128 + w) * 128 + h) * 128;
  float* rowW = attW + (((size_t)b * 128 + h) * 128 + w) * 128;

  float4 hv = ((float4*)rowH)[l];
  float4 wf = ((float4*)rowW)[l];

  float m = fmaxf(fmaxf(fmaxf(hv.x, hv.y), fmaxf(hv.z, hv.w)),
                  fmaxf(fmaxf(wf.x, wf.y), fmaxf(wf.z, wf.w)));
#pragma unroll
  for (int s = 16; s > 0; s >>= 1) m = fmaxf(m, __shfl_xor(m, s, 32));

  hv.x = __expf(hv.x - m); hv.y = __expf(hv.y - m);
  hv.z = __expf(hv.z - m); hv.w = __expf(hv.w - m);
  wf.x = __expf(wf.x - m); wf.y = __expf(wf.y - m);
  wf.z = __expf(wf.z - m); wf.w = __expf(wf.w - m);

  float s = hv.x + hv.y + hv.z + hv.w + wf.x + wf.y + wf.z + wf.w;
#pragma unroll
  for (int i = 16; i > 0; i >>= 1) s += __shfl_xor(s, i, 32);
  float inv = 1.f / s;

  hv.x *= inv; hv.y *= inv; hv.z *= inv; hv.w *= inv;
  wf.x *= inv; wf.y *= inv; wf.z *= inv; wf.w *= inv;
  ((float4*)rowH)[l] = hv;
  ((float4*)rowW)[l] = wf;
}

// ---------------------------------------------------------------------------
// Kernel D: attention apply.  One block per (b, x, n-half) computes
// Out[c][n] = sum_j V(c,j) * Att[x][n][j], 256(M) x 64(N), K=128.
// V element: (b*C+c)*HW + x*sx + j*sp;  output: (b*C+c)*HW + x*sx + n*sp.
// Pass 1 (att_H, sx=1,sp=W): out = gamma*acc.
// Pass 2 (att_W, sx=W,sp=1): out += gamma*acc + mask  (final result).
// att tile staged with async b128 copies; pad 36 => 16B rows + conflict-free
// transposed v2f reads.
// ---------------------------------------------------------------------------
__global__ __launch_bounds__(256) void outproj_kernel(
    const float* __restrict__ v, const float* __restrict__ att,
    const float* __restrict__ maskf, float* __restrict__ out,
    const float* __restrict__ gamma, int sx, int sp, int accumulate)
{
  __shared__ float as[64][36];

  const int t  = threadIdx.x;
  const int x  = blockIdx.x >> 1;
  const int nh = (blockIdx.x & 1) * 64;
  const int b  = blockIdx.y;
  const int wv = t >> 5;
  const int tn = t & 15;
  const int hh = (t >> 4) & 1;
  const int wm = (wv >> 1) * 64;           // M offset: 0/64/128/192
  const int wnl = (wv & 1) * 32;           // local N offset within 64-tile
  const float g = gamma[0];
  const size_t ab   = ((size_t)b * 128 + x) * 16384;
  const size_t xoff = (size_t)x * sx;

  const v8f zero = {0.f, 0.f, 0.f, 0.f, 0.f, 0.f, 0.f, 0.f};
  v8f acc[4][2];
#pragma unroll
  for (int i = 0; i < 4; i++) { acc[i][0] = zero; acc[i][1] = zero; }

  for (int j0 = 0; j0 < 128; j0 += 32) {
    __syncthreads();
#pragma unroll
    for (int i = 0; i < 2; i++) {
      int idx4 = i * 256 + t;          // 512 float4 tiles
      int nn = idx4 >> 3;              // 0..63
      int kk = (idx4 & 7) * 4;         // 0,4,...,28
      async_ld_b128(att + ab + (size_t)(nh + nn) * 128 + j0 + kk, &as[nn][kk]);
    }
    wait_async0();
    __syncthreads();

    for (int k = 0; k < 32; k += 4) {
      v2f a[4], bb[2];
#pragma unroll
      for (int mt = 0; mt < 4; mt++) {
        int m = wm + mt * 16 + tn;
        size_t base = (size_t)(b * Cc + m) * HWc + xoff;
        a[mt].x = v[base + (size_t)(j0 + k + 2 * hh) * sp];
        a[mt].y = v[base + (size_t)(j0 + k + 2 * hh + 1) * sp];
      }
#pragma unroll
      for (int nt = 0; nt < 2; nt++) {
        int n = wnl + nt * 16 + tn;
        bb[nt] = *(const v2f*)&as[n][k + 2 * hh];
      }
#pragma unroll
      for (int mt = 0; mt < 4; mt++)
#pragma unroll
        for (int nt = 0; nt < 2; nt++)
          acc[mt][nt] = WMMA_F32(a[mt], bb[nt], acc[mt][nt]);
    }
  }

#pragma unroll
  for (int mt = 0; mt < 4; mt++)
#pragma unroll
    for (int nt = 0; nt < 2; nt++)
#pragma unroll
      for (int r = 0; r < 8; r++) {
        int row = wm + mt * 16 + r + 8 * hh;
        int col = nh + wnl + nt * 16 + tn;
        size_t o = (size_t)(b * Cc + row) * HWc + xoff + (size_t)col * sp;
        float val = g * acc[mt][nt][r];
        out[o] = accumulate ? (out[o] + val + maskf[o]) : val;
      }
}

// ---------------------------------------------------------------------------
extern "C" void kernel_launch(void* const* d_in, const int* in_sizes, int n_in,
                              void* d_out, int out_size, void* d_ws, size_t ws_size,
                              hipStream_t stream)
{
  (void)in_sizes; (void)n_in; (void)out_size; (void)ws_size;

  const float* edge  = (const float*)d_in[0];
  const float* maskf = (const float*)d_in[1];
  const float* Wq    = (const float*)d_in[2];
  const float* bq    = (const float*)d_in[3];
  const float* Wk    = (const float*)d_in[4];
  const float* bk    = (const float*)d_in[5];
  const float* Wv    = (const float*)d_in[6];
  const float* bv    = (const float*)d_in[7];
  const float* gamma = (const float*)d_in[8];

  float* out  = (float*)d_out;
  float* attH = out + OUT_N;
  float* attW = attH + ATTH_N;

  float* q = (float*)d_ws;
  float* k = q + OUT_N;
  float* v = k + OUT_N;

  // 1) QKV projection (GEMM, WMMA f32)
  qkv_kernel<<<dim3(HWc / 32, Bc), 256, 0, stream>>>(
      edge, maskf, Wq, bq, Wk, bk, Wv, bv, q, k, v);

  // 2) Raw energies straight into the attention output regions
  energy_kernel<<<dim3(Wc, Bc), 256, 0, stream>>>(q, k, attH, 1, Wc);  // E_H
  energy_kernel<<<dim3(Hc, Bc), 256, 0, stream>>>(q, k, attW, Wc, 1);  // E_W

  // 3) Joint softmax over [E_H row ; E_W row], in place
  softmax_kernel<<<(Bc * Hc * Wc) / 8, 256, 0, stream>>>(attH, attW);

  // 4) Attention apply: out = gamma*out_H, then out += gamma*out_W + mask
  outproj_kernel<<<dim3(Wc * 2, Bc), 256, 0, stream>>>(
      v, attH, maskf, out, gamma, 1, Wc, 0);
  outproj_kernel<<<dim3(Hc * 2, Bc), 256, 0, stream>>>(
      v, attW, maskf, out, gamma, Wc, 1, 1);
}